// LanguageModelCriterion_82102594831150
// MI455X (gfx1250) — compile-verified
//
#include <hip/hip_runtime.h>
#include <hip/hip_bf16.h>

typedef __attribute__((ext_vector_type(2))) float v2f;
typedef __attribute__((ext_vector_type(8))) float v8f;

#define WAVE 32
#define WAVES_PER_BLOCK 8

// One wave32 per (b,s) row: coalesced int4/float4 loads cover C candidates,
// per-lane "last match wins" scan, then a 5-step shuffle arg-max over the
// candidate index (largest index wins == JAX scatter last-write semantics).
// Per-block partial sums (loss, mask) go to workspace — no float atomics, so
// the result is bit-deterministic across graph replays. Blocks past the last
// row write 0.0 partials, zero-padding the workspace up to a multiple of 64
// entries so the finalize WMMA loop needs no bounds checks.
__global__ void lmcrit_partials(const float* __restrict__ logits,
                                const int*   __restrict__ target,
                                const int*   __restrict__ rel_candidate,
                                const int*   __restrict__ target_num,
                                const int*   __restrict__ rel_candidate_num,
                                const float* __restrict__ mask,
                                const float* __restrict__ rel_weight,
                                float* __restrict__ wsLoss,
                                float* __restrict__ wsMask,
                                int S, int C, int rows) {
    __shared__ float ldsL[WAVES_PER_BLOCK];
    __shared__ float ldsM[WAVES_PER_BLOCK];
    const int w    = threadIdx.x >> 5;
    const int lane = threadIdx.x & 31;
    const int row  = blockIdx.x * WAVES_PER_BLOCK + w;

    float contrib = 0.0f, mval = 0.0f;
    if (row < rows) {
        const int b  = row / S;
        const int s  = row - b * S;
        const int t  = target[row];
        const int tn = target_num[b];
        const int cn = rel_candidate_num[row];

        int   bestC = -1;
        float bestV = 0.0f;
        if (s < tn) {
            const int*   crow = rel_candidate + (size_t)row * C;
            const float* lrow = logits        + (size_t)row * C;
            // C is a multiple of 4 (100); 16B-aligned vector loads.
            for (int c0 = lane * 4; c0 < C; c0 += 4 * WAVE) {
                int4   cc = *(const int4*)  (crow + c0);
                float4 ll = *(const float4*)(lrow + c0);
                // ascending c within lane -> later match overwrites (last wins)
                if (c0 + 0 < cn && cc.x == t) { bestC = c0 + 0; bestV = ll.x; }
                if (c0 + 1 < cn && cc.y == t) { bestC = c0 + 1; bestV = ll.y; }
                if (c0 + 2 < cn && cc.z == t) { bestC = c0 + 2; bestV = ll.z; }
                if (c0 + 3 < cn && cc.w == t) { bestC = c0 + 3; bestV = ll.w; }
            }
        }
        // wave32 arg-max over candidate index
        #pragma unroll
        for (int off = 16; off >= 1; off >>= 1) {
            int   oc = __shfl_xor(bestC, off, WAVE);
            float ov = __shfl_xor(bestV, off, WAVE);
            if (oc > bestC) { bestC = oc; bestV = ov; }
        }
        contrib = (bestC >= 0) ? (-rel_weight[row] * bestV) : 0.0f;
        mval    = mask[row];
    }
    if (lane == 0) { ldsL[w] = contrib; ldsM[w] = mval; }
    __syncthreads();
    if (threadIdx.x == 0) {
        float sL = 0.0f, sM = 0.0f;
        #pragma unroll
        for (int i = 0; i < WAVES_PER_BLOCK; ++i) { sL += ldsL[i]; sM += ldsM[i]; }
        wsLoss[blockIdx.x] = sL;
        wsMask[blockIdx.x] = sM;
    }
}

// Single-wave finalize: reduce the per-block partials with
// V_WMMA_F32_16X16X4_F32 using B = ones (D[m][n] = rowsum(A[m])), chaining C
// across chunks of 64 values. A-fragment layout (ISA 16x4 f32): lanes 0-15
// hold K=0,1; lanes 16-31 hold K=2,3 -> per-lane offset 4*(lane&15)+2*(lane>>4),
// loaded as one contiguous v2f (global_load_b64). nPart is a multiple of 64
// and the producer zero-pads, so the loop is branch-free and EXEC stays all-1s.
// Fixed-order accumulation -> deterministic across graph replays.
__global__ void lmcrit_finalize(const float* __restrict__ wsLoss,
                                const float* __restrict__ wsMask,
                                float* __restrict__ out, int nPart) {
    const int lane = threadIdx.x & 31;
    const int off  = 4 * (lane & 15) + 2 * (lane >> 4);

    v2f bOnes; bOnes.x = 1.0f; bOnes.y = 1.0f;
    v8f cL = {}; v8f cM = {};

    const int chunks = nPart >> 6;          // nPart % 64 == 0 by construction
    const float* pL = wsLoss + off;
    const float* pM = wsMask + off;
    for (int i = 0; i < chunks; ++i) {
        v2f aL = *(const v2f*)(pL);         // one b64 load per matrix per chunk
        v2f aM = *(const v2f*)(pM);
        cL = __builtin_amdgcn_wmma_f32_16x16x4_f32(false, aL, false, bOnes,
                                                   (short)0, cL, false, false);
        cM = __builtin_amdgcn_wmma_f32_16x16x4_f32(false, aM, false, bOnes,
                                                   (short)0, cM, false, false);
        pL += 64;
        pM += 64;
    }
    // D column 0: rows 0-7 live in lane 0's 8 VGPRs, rows 8-15 in lane 16's.
    float sL = 0.0f, sM = 0.0f;
    #pragma unroll
    for (int r = 0; r < 8; ++r) { sL += cL[r]; sM += cM[r]; }
    sL += __shfl_xor(sL, 16, WAVE);
    sM += __shfl_xor(sM, 16, WAVE);
    if (lane == 0) out[0] = sL / sM;
}

extern "C" void kernel_launch(void* const* d_in, const int* in_sizes, int n_in,
                              void* d_out, int out_size, void* d_ws, size_t ws_size,
                              hipStream_t stream) {
    (void)n_in; (void)out_size; (void)ws_size;
    const float* logits            = (const float*)d_in[0];
    const int*   target            = (const int*)  d_in[1];
    const int*   rel_candidate     = (const int*)  d_in[2];
    /* d_in[3] = rel_wordlist_num: dummy column V is dropped -> unused */
    const int*   target_num        = (const int*)  d_in[4];
    const int*   rel_candidate_num = (const int*)  d_in[5];
    const float* mask              = (const float*)d_in[6];
    const float* rel_weight        = (const float*)d_in[7];
    float*       out               = (float*)d_out;

    const int N    = in_sizes[4];                  // 64
    const int S    = in_sizes[1] / N;              // 50
    const int C    = in_sizes[0] / in_sizes[1];    // 100
    const int rows = N * S;                        // 3200
    const int nBlocks    = (rows + WAVES_PER_BLOCK - 1) / WAVES_PER_BLOCK;  // 400
    const int nBlocksPad = (nBlocks + 63) & ~63;                            // 448

    float* wsLoss = (float*)d_ws;
    float* wsMask = wsLoss + nBlocksPad;

    // Padded grid: trailing blocks (all rows >= rows) write 0.0 partials,
    // so wsLoss/wsMask[0..nBlocksPad) are fully initialized every call.
    lmcrit_partials<<<nBlocksPad, WAVES_PER_BLOCK * WAVE, 0, stream>>>(
        logits, target, rel_candidate, target_num, rel_candidate_num,
        mask, rel_weight, wsLoss, wsMask, S, C, rows);

    lmcrit_finalize<<<1, WAVE, 0, stream>>>(wsLoss, wsMask, out, nBlocksPad);
}